// Fastformer_5669356831990
// MI455X (gfx1250) — compile-verified
//
#include <hip/hip_runtime.h>

typedef __attribute__((ext_vector_type(2))) float v2f;
typedef __attribute__((ext_vector_type(8))) float v8f;
typedef __attribute__((ext_vector_type(4))) int   v4i;

#define B_     8
#define N_     8192
#define DIM_   256
#define D_     64
#define NTILE  64                         // rows per workgroup
#define TPB    (N_ / NTILE)               // 128 tiles per batch
#define NBLK   (B_ * TPB)                 // 1024 blocks
#define SCALE_ 0.125f                     // 64^-0.5
#define MASKV  (-3.402823466e38f)

#define AS1 __attribute__((address_space(1)))
#define AS3 __attribute__((address_space(3)))

// ---- gfx1250 async global->LDS copy (ASYNCcnt path), with safe fallback ----
// builtin signature: (v4i AS1* global_src, v4i AS3* lds_dst, imm offset, imm cpol)
__device__ __forceinline__ void async_copy16(const void* g, void* l)
{
#if __has_builtin(__builtin_amdgcn_global_load_async_to_lds_b128)
    __builtin_amdgcn_global_load_async_to_lds_b128(
        (AS1 v4i*)g, (AS3 v4i*)l, 0, 0);
#else
    *(float4*)l = *(const float4*)g;      // synchronous fallback
#endif
}

__device__ __forceinline__ void wait_async()
{
#if __has_builtin(__builtin_amdgcn_s_wait_asynccnt)
    __builtin_amdgcn_s_wait_asynccnt(0);
#else
    asm volatile("s_wait_asynccnt 0" ::: "memory");
#endif
}

// ---- kernel-1 dynamic-LDS layout (float offsets) ----
#define XPITCH 260                        // 1040B rows: 16B aligned, lane stride 4 mod 64 banks
#define QPITCH 65                         // scalar ds ops only: conflict-free both passes
#define XOFF   0
#define QOFF   (NTILE * XPITCH)           // 16640
#define RMAXO  (QOFF + NTILE * QPITCH)    // 20800
#define RINVO  (RMAXO + NTILE)
#define RMSKO  (RINVO + NTILE)
#define K1_LDS_FLOATS (RMSKO + NTILE)     // 20992 floats = 83968 B (< gfx1250 320KB/WGP)

// ============================================================================
// Phase 1: q/k/v GEMMs (fp32 WMMA) + alpha softmax + partial global_q
// ============================================================================
__global__ void __launch_bounds__(128)
k_qkv(const float* __restrict__ x, const unsigned char* __restrict__ mask,
      const float* __restrict__ Wq, const float* __restrict__ Wk,
      const float* __restrict__ Wv, const float* __restrict__ alpha,
      float* __restrict__ q, float* __restrict__ k, float* __restrict__ v,
      float* __restrict__ pgq)
{
    extern __shared__ float lds[];
    const int tid  = threadIdx.x;
    const int wave = tid >> 5;
    const int lane = tid & 31;
    const int b    = blockIdx.x >> 7;               // /TPB
    const int n0   = (blockIdx.x & (TPB - 1)) * NTILE;
    const size_t rowG = (size_t)b * N_ + n0;

    // ---- async-stage 64x256 x-tile straight into LDS (no VGPR round-trip) ----
    {
        const float* xs = x + rowG * DIM_;
        for (int i = tid; i < NTILE * (DIM_ / 4); i += 128) {
            const int r = i >> 6, c4 = i & 63;
            async_copy16(xs + (size_t)r * DIM_ + (c4 << 2),
                         &lds[XOFF + r * XPITCH + (c4 << 2)]);
        }
    }
    wait_async();
    __syncthreads();

    // ---- WMMA: 16 rows per wave, 4 column tiles, K=256 in steps of 4 ----
    const int m  = lane & 15;            // A row / B column within tile
    const int kk = (lane >> 4) << 1;     // K sub-offset {0,2}
    const int rowBase = wave * 16;
    const float* srcA = &lds[XOFF + (rowBase + m) * XPITCH];

    for (int nt = 0; nt < 4; ++nt) {
        v8f cq = {}, ck = {}, cv = {};
        const int nCol = (nt << 4) + m;
#pragma unroll 4
        for (int ks = 0; ks < DIM_ / 4; ++ks) {
            const int kb = (ks << 2) + kk;
            v2f a, bq, bk, bv;
            a[0]  = srcA[kb];                a[1]  = srcA[kb + 1];
            bq[0] = Wq[kb * D_ + nCol];      bq[1] = Wq[(kb + 1) * D_ + nCol];
            bk[0] = Wk[kb * D_ + nCol];      bk[1] = Wk[(kb + 1) * D_ + nCol];
            bv[0] = Wv[kb * D_ + nCol];      bv[1] = Wv[(kb + 1) * D_ + nCol];
            cq = __builtin_amdgcn_wmma_f32_16x16x4_f32(false, a, false, bq, (short)0, cq, false, false);
            ck = __builtin_amdgcn_wmma_f32_16x16x4_f32(false, a, false, bk, (short)0, ck, false, false);
            cv = __builtin_amdgcn_wmma_f32_16x16x4_f32(false, a, false, bv, (short)0, cv, false, false);
        }
        // C/D layout: VGPR g -> row g + 8*(lane>>4), col = lane&15
#pragma unroll
        for (int g = 0; g < 8; ++g) {
            const int row = rowBase + g + ((lane >> 4) << 3);
            const int col = (nt << 4) + m;
            const size_t off = (rowG + row) * D_ + col;
            q[off] = cq[g];
            k[off] = ck[g];
            v[off] = cv[g];
            lds[QOFF + row * QPITCH + col] = cq[g];   // q tile for the softmax phase
        }
    }
    __syncthreads();

    // ---- per-row alpha-softmax stats (row r handled by thread r) ----
    if (tid < NTILE) {
        const int r = tid;
        const float mk = (mask[rowG + r] != 0) ? 1.0f : 0.0f;
        float mx = MASKV;
        for (int d = 0; d < D_; ++d) {
            const float lg = (mk != 0.0f) ? lds[QOFF + r * QPITCH + d] * alpha[d] * SCALE_
                                          : MASKV;
            mx = fmaxf(mx, lg);
        }
        float s = 0.0f;
        for (int d = 0; d < D_; ++d) {
            const float lg = (mk != 0.0f) ? lds[QOFF + r * QPITCH + d] * alpha[d] * SCALE_
                                          : MASKV;
            s += __expf(lg - mx);
        }
        lds[RMAXO + r] = mx;
        lds[RINVO + r] = 1.0f / s;
        lds[RMSKO + r] = mk;
    }
    __syncthreads();

    // ---- column pass: partial global_q[d] over this block's 64 rows ----
    if (tid < D_) {
        const int d = tid;
        const float ad = alpha[d] * SCALE_;
        float acc = 0.0f;
        for (int r = 0; r < NTILE; ++r) {
            const float qv = lds[QOFF + r * QPITCH + d];
            const float lg = (lds[RMSKO + r] != 0.0f) ? qv * ad : MASKV;
            acc += qv * __expf(lg - lds[RMAXO + r]) * lds[RINVO + r];
        }
        pgq[(size_t)blockIdx.x * D_ + d] = acc;      // deterministic: no atomics
    }
}

// ============================================================================
// Deterministic reduction of per-block partials: dst[b*64+d] = sum_t part[...]
// ============================================================================
__global__ void __launch_bounds__(512)
k_reduce(const float* __restrict__ part, float* __restrict__ dst)
{
    const int t = threadIdx.x;          // 512 = B_*D_
    const int b = t >> 6, d = t & 63;
    float s = 0.0f;
    for (int i = 0; i < TPB; ++i)
        s += part[((size_t)(b * TPB + i)) * D_ + d];
    dst[t] = s;
}

// ============================================================================
// Phase 2: p = global_q (.) k, beta softmax, partial global_k
// ============================================================================
#define KP 68   // 272B rows: 16B aligned for async B128 writes
__global__ void __launch_bounds__(128)
k_beta(const float* __restrict__ kmat, const unsigned char* __restrict__ mask,
       const float* __restrict__ beta, const float* __restrict__ gq,
       float* __restrict__ pgk)
{
    __shared__ float sp[NTILE * KP];
    __shared__ float sgq[D_];
    __shared__ float rmax[NTILE], rinv[NTILE], rmsk[NTILE];
    const int tid = threadIdx.x;
    const int b   = blockIdx.x >> 7;
    const int n0  = (blockIdx.x & (TPB - 1)) * NTILE;
    const size_t rowG = (size_t)b * N_ + n0;

    // async-stage raw k tile; gq staged separately (scaling applied on read)
    const float* ks0 = kmat + rowG * D_;
    for (int i = tid; i < NTILE * (D_ / 4); i += 128) {
        const int r = i >> 4, c4 = i & 15;
        async_copy16(ks0 + (size_t)r * D_ + (c4 << 2), &sp[r * KP + (c4 << 2)]);
    }
    if (tid < D_) sgq[tid] = gq[b * D_ + tid];
    wait_async();
    __syncthreads();

    if (tid < NTILE) {
        const int r = tid;
        const float mk = (mask[rowG + r] != 0) ? 1.0f : 0.0f;
        float mx = MASKV;
        for (int d = 0; d < D_; ++d) {
            const float p  = sp[r * KP + d] * sgq[d];
            const float lg = (mk != 0.0f) ? p * beta[d] * SCALE_ : MASKV;
            mx = fmaxf(mx, lg);
        }
        float s = 0.0f;
        for (int d = 0; d < D_; ++d) {
            const float p  = sp[r * KP + d] * sgq[d];
            const float lg = (mk != 0.0f) ? p * beta[d] * SCALE_ : MASKV;
            s += __expf(lg - mx);
        }
        rmax[r] = mx; rinv[r] = 1.0f / s; rmsk[r] = mk;
    }
    __syncthreads();

    if (tid < D_) {
        const int d = tid;
        const float gd = sgq[d];
        const float bd = beta[d] * SCALE_;
        float acc = 0.0f;
        for (int r = 0; r < NTILE; ++r) {
            const float p  = sp[r * KP + d] * gd;
            const float lg = (rmsk[r] != 0.0f) ? p * bd : MASKV;
            acc += p * __expf(lg - rmax[r]) * rinv[r];
        }
        pgk[(size_t)blockIdx.x * D_ + d] = acc;
    }
}

// ============================================================================
// Phase 3: out = (global_k (.) v) @ Wr + q   (fp32 WMMA, K=64)
// ============================================================================
#define VPITCH 68   // 272B rows: 16B aligned; lane stride 4 mod 64 banks for frags
__global__ void __launch_bounds__(128)
k_out(const float* __restrict__ vmat, const float* __restrict__ qmat,
      const float* __restrict__ Wr, const float* __restrict__ gk,
      float* __restrict__ out)
{
    __shared__ float sv[NTILE * VPITCH];
    __shared__ float sgk[D_];
    const int tid = threadIdx.x, wave = tid >> 5, lane = tid & 31;
    const int b  = blockIdx.x >> 7;
    const int n0 = (blockIdx.x & (TPB - 1)) * NTILE;
    const size_t rowG = (size_t)b * N_ + n0;

    // async-stage raw v tile + gk, then one in-place LDS scale pass: sv = gk .* v
    const float* vs = vmat + rowG * D_;
    for (int i = tid; i < NTILE * (D_ / 4); i += 128) {
        const int r = i >> 4, c4 = i & 15;
        async_copy16(vs + (size_t)r * D_ + (c4 << 2), &sv[r * VPITCH + (c4 << 2)]);
    }
    if (tid < D_) sgk[tid] = gk[b * D_ + tid];
    wait_async();
    __syncthreads();

    for (int i = tid; i < NTILE * (D_ / 4); i += 128) {
        const int r = i >> 4, c4 = i & 15;
        float4* p = (float4*)&sv[r * VPITCH + (c4 << 2)];
        float4 t = *p;
        const float4 g = ((const float4*)sgk)[c4];
        t.x *= g.x; t.y *= g.y; t.z *= g.z; t.w *= g.w;
        *p = t;
    }
    __syncthreads();

    const int m = lane & 15, kk = (lane >> 4) << 1, rowBase = wave * 16;
    const float* srcA = &sv[(rowBase + m) * VPITCH];

    for (int nt = 0; nt < 4; ++nt) {
        v8f c = {};
        const int nCol = (nt << 4) + m;
#pragma unroll
        for (int ksi = 0; ksi < D_ / 4; ++ksi) {
            const int kb = (ksi << 2) + kk;
            v2f a, bw;
            a[0]  = srcA[kb];              a[1]  = srcA[kb + 1];
            bw[0] = Wr[kb * D_ + nCol];    bw[1] = Wr[(kb + 1) * D_ + nCol];
            c = __builtin_amdgcn_wmma_f32_16x16x4_f32(false, a, false, bw, (short)0, c, false, false);
        }
#pragma unroll
        for (int g = 0; g < 8; ++g) {
            const int row = rowBase + g + ((lane >> 4) << 3);
            const int col = (nt << 4) + m;
            const size_t off = (rowG + row) * D_ + col;
            out[off] = c[g] + qmat[off];   // fused +q residual
        }
    }
}

// ============================================================================
// Host-side launcher
// ============================================================================
extern "C" void kernel_launch(void* const* d_in, const int* in_sizes, int n_in,
                              void* d_out, int out_size, void* d_ws, size_t ws_size,
                              hipStream_t stream)
{
    (void)in_sizes; (void)n_in; (void)out_size; (void)ws_size;

    const float*         x     = (const float*)d_in[0];
    const unsigned char* mask  = (const unsigned char*)d_in[1];
    const float*         Wq    = (const float*)d_in[2];
    const float*         Wk    = (const float*)d_in[3];
    const float*         Wv    = (const float*)d_in[4];
    const float*         Wr    = (const float*)d_in[5];
    const float*         alpha = (const float*)d_in[6];
    const float*         beta  = (const float*)d_in[7];
    float*               outp  = (float*)d_out;

    // workspace layout (floats): q | k | v | pgq | pgk | gq | gk   (~48.5 MB)
    const size_t QKV = (size_t)B_ * N_ * D_;        // 4,194,304
    float* ws  = (float*)d_ws;
    float* q   = ws;
    float* k   = ws + QKV;
    float* v   = ws + 2 * QKV;
    float* pgq = ws + 3 * QKV;                      // NBLK * 64
    float* pgk = pgq + (size_t)NBLK * D_;
    float* gq  = pgk + (size_t)NBLK * D_;
    float* gk  = gq + B_ * D_;

    k_qkv<<<NBLK, 128, K1_LDS_FLOATS * sizeof(float), stream>>>(
        x, mask, Wq, Wk, Wv, alpha, q, k, v, pgq);
    k_reduce<<<1, 512, 0, stream>>>(pgq, gq);
    k_beta<<<NBLK, 128, 0, stream>>>(k, mask, beta, gq, pgk);
    k_reduce<<<1, 512, 0, stream>>>(pgk, gk);
    k_out<<<NBLK, 128, 0, stream>>>(v, q, Wr, gk, outp);
}